// INRF_5970004541758
// MI455X (gfx1250) — compile-verified
//
#include <hip/hip_runtime.h>

typedef __attribute__((ext_vector_type(2))) float v2f;
typedef __attribute__((ext_vector_type(8))) float v8f;

#define H   32
#define W   32
#define CIN 3
#define F   16
#define KK  75            // 5*5*3 im2col depth
#define PIX (H * W)       // 1024 pixels per image
#define MPB (PIX * CIN)   // 3072  (x elements per batch)
#define NPB (PIX * F)     // 16384 (blurred elements per batch)

// ---------------------------------------------------------------------------
// Branchless im2col gather with SAME padding: clamp to an always-valid
// address, load unconditionally, v_cndmask the result. No EXEC divergence.
// ---------------------------------------------------------------------------
__device__ __forceinline__ float patch_val(const float* __restrict__ src,
                                           int b, int pix, int k) {
    int kc  = (k < KK) ? k : 0;        // K padded 75 -> 76
    int p   = kc / 15;
    int rem = kc - p * 15;
    int q   = rem / 3;
    int c   = rem - q * 3;
    int py  = pix >> 5;
    int px  = pix & 31;
    int i   = py + p - 2;
    int j   = px + q - 2;
    bool ok = (k < KK) & (i >= 0) & (i < H) & (j >= 0) & (j < W);
    int ic  = min(max(i, 0), H - 1);
    int jc  = min(max(j, 0), W - 1);
    float v = src[((b * H + ic) * W + jc) * CIN + c];
    return ok ? v : 0.0f;
}

// ---------------------------------------------------------------------------
// 5x5 conv (C=3 -> F=16) as GEMM on V_WMMA_F32_16X16X4_F32 (fp32-exact).
// One wave per 16-pixel tile: D[16 pix x 16 filt] = A[16x76] * B[76x16].
// Fully unrolled K loop -> 19 (or 38 dual) straight-line WMMAs.
// Optional second filter bank (w2/d2) reuses the A fetch (first & blurred).
// Optional base/sign fuses   d1 = base + sign * conv(src, w1).
// All remaining branches are wave-uniform -> EXEC all-1s at every WMMA.
// ---------------------------------------------------------------------------
__global__ void conv5x5_wmma(const float* __restrict__ src,
                             const float* __restrict__ w1, float* __restrict__ d1,
                             const float* __restrict__ w2, float* __restrict__ d2,
                             const float* __restrict__ base, float sign) {
    int lane  = threadIdx.x & 31;
    int tile  = (blockIdx.x * blockDim.x + threadIdx.x) >> 5;  // wave id
    int b     = tile >> 6;            // 64 tiles per batch image
    int pbase = (tile & 63) << 4;     // first pixel of this 16-pixel tile
    int mrow  = lane & 15;            // A row M / B col N for this lane
    int koff  = (lane >> 4) << 1;     // lanes 0-15 -> K+{0,1}; 16-31 -> K+{2,3}

    v8f acc1 = {};
    v8f acc2 = {};
    #pragma unroll
    for (int k0 = 0; k0 < 76; k0 += 4) {
        int ka = k0 + koff;
        v2f a;
        a.x = patch_val(src, b, pbase + mrow, ka);
        a.y = patch_val(src, b, pbase + mrow, ka + 1);
        // weights are [5,5,3,16] -> B[k][n] = w[k*16 + n]; branchless pad
        int kb0 = (ka     < KK) ? ka     : 0;
        int kb1 = (ka + 1 < KK) ? ka + 1 : 0;
        v2f bv;
        bv.x = (ka     < KK) ? w1[kb0 * F + mrow] : 0.0f;
        bv.y = (ka + 1 < KK) ? w1[kb1 * F + mrow] : 0.0f;
        acc1 = __builtin_amdgcn_wmma_f32_16x16x4_f32(
            false, a, false, bv, (short)0, acc1, false, false);
        if (w2) {
            v2f cv;
            cv.x = (ka     < KK) ? w2[kb0 * F + mrow] : 0.0f;
            cv.y = (ka + 1 < KK) ? w2[kb1 * F + mrow] : 0.0f;
            acc2 = __builtin_amdgcn_wmma_f32_16x16x4_f32(
                false, a, false, cv, (short)0, acc2, false, false);
        }
    }

    // D layout: VGPR r -> M = r (lanes 0-15) or r+8 (lanes 16-31); N = lane%16
    int col = mrow;
    int rb  = (lane >> 4) << 3;
    #pragma unroll
    for (int r = 0; r < 8; ++r) {
        int pix = pbase + rb + r;
        int idx = (b * PIX + pix) * F + col;
        float v = acc1[r];
        d1[idx] = base ? (base[idx] + sign * v) : (sign * v);
        if (w2) d2[idx] = acc2[r];
    }
}

// ---------------------------------------------------------------------------
// difference[b,m] = sum_n tanh(x[b,m] - blurred[b,n])  -- the 100M-tanh loop.
// One batch / 256 m-values / one 1024-wide n-slice per block; slice staged in
// LDS (broadcast reads, conflict-free, ds_load_b128). Hardware v_tanh_f32
// (CDNA5 TRANS op) batched 4-per-asm-block: every tanh is followed by >=1
// independent instruction (next tanh or the single trailing v_nop), which
// satisfies the ISA's TRANS result hazard without per-op nops.
// ---------------------------------------------------------------------------
#define SLICE   1024
#define NSLICES (NPB / SLICE)   // 16
#define MT      (MPB / 256)     // 12

__global__ void tanh_diff(const float* __restrict__ x,
                          const float* __restrict__ blurred,
                          float* __restrict__ diff) {
    __shared__ float ys[SLICE];
    int bid = blockIdx.x;
    int s   = bid & (NSLICES - 1);
    int mt  = (bid >> 4) % MT;
    int b   = bid / (NSLICES * MT);

    const float* yb = blurred + b * NPB + s * SLICE;
    for (int i = threadIdx.x; i < SLICE; i += blockDim.x) ys[i] = yb[i];
    __syncthreads();

    int m     = mt * 256 + threadIdx.x;
    float xv  = x[b * MPB + m];
    float acc0 = 0.0f, acc1 = 0.0f;
    const float4* ys4 = reinterpret_cast<const float4*>(ys);
    #pragma unroll 4
    for (int i = 0; i < SLICE / 4; ++i) {
        float4 y = ys4[i];                       // ds_load_b128, broadcast
        float t0 = xv - y.x, t1 = xv - y.y, t2 = xv - y.z, t3 = xv - y.w;
        float r0, r1, r2, r3;
        asm volatile("v_tanh_f32 %0, %4\n\t"
                     "v_tanh_f32 %1, %5\n\t"
                     "v_tanh_f32 %2, %6\n\t"
                     "v_tanh_f32 %3, %7\n\t"
                     "v_nop"
                     : "=&v"(r0), "=&v"(r1), "=&v"(r2), "=&v"(r3)
                     : "v"(t0), "v"(t1), "v"(t2), "v"(t3));
        acc0 += r0 + r2;
        acc1 += r1 + r3;
    }
    atomicAdd(&diff[b * MPB + m], acc0 + acc1);
}

__global__ void zero_f32(float* __restrict__ p, int n) {
    int i = blockIdx.x * blockDim.x + threadIdx.x;
    if (i < n) p[i] = 0.0f;
}

// ---------------------------------------------------------------------------
// launch: zero diff -> dual conv (first+blurred) -> tanh reduction ->
//         final conv fused with (first - second)
// ---------------------------------------------------------------------------
extern "C" void kernel_launch(void* const* d_in, const int* in_sizes, int n_in,
                              void* d_out, int out_size, void* d_ws, size_t ws_size,
                              hipStream_t stream) {
    const float* x = (const float*)d_in[0];
    const float* m = (const float*)d_in[1];
    const float* w = (const float*)d_in[2];
    const float* g = (const float*)d_in[3];
    float* out = (float*)d_out;
    int B = in_sizes[0] / MPB;    // 2

    float* first   = (float*)d_ws;              // B*NPB floats
    float* blurred = first + (size_t)B * NPB;   // B*NPB floats
    float* diff    = blurred + (size_t)B * NPB; // B*MPB floats

    zero_f32<<<(B * MPB + 255) / 256, 256, 0, stream>>>(diff, B * MPB);

    int tiles = B * (PIX / 16);   // 128 waves, 4 waves/block
    conv5x5_wmma<<<tiles / 4, 128, 0, stream>>>(x, m, first, g, blurred,
                                                nullptr, 1.0f);

    tanh_diff<<<B * NSLICES * MT, 256, 0, stream>>>(x, blurred, diff);

    conv5x5_wmma<<<tiles / 4, 128, 0, stream>>>(diff, w, out, nullptr, nullptr,
                                                first, -1.0f);
}